// ChunkLayer_30983894073943
// MI455X (gfx1250) — compile-verified
//
#include <hip/hip_runtime.h>

// Problem constants from the reference: B=4, L=8192, D=1024.
#define BB 4
#define LL 8192
#define DD 1024
#define ROWS_PER_ITER 4                 // rows moved per block-iteration
#define ROW_BYTES (DD * 4)              // 4 KB per row

// ---------------------------------------------------------------------------
// Kernel 1: per-batch stable compaction ranks via block scan, scatter perm[]
//   perm[b][j] = source row index for output row j  (all L entries written;
//   the gather only consumes the first S of them).
//   numtok[b]  = number of selected tokens in batch b.
// One 1024-thread block per batch; each thread owns 8 contiguous positions.
// ---------------------------------------------------------------------------
__global__ __launch_bounds__(1024)
void ChunkLayer_scan_scatter(const int* __restrict__ mask,
                             int* __restrict__ perm,
                             int* __restrict__ numtok) {
    __shared__ int s[1024];
    const int b = blockIdx.x;
    const int t = threadIdx.x;
    const int* m = mask + (size_t)b * LL;

    int loc[8];
    int c = 0;
#pragma unroll
    for (int i = 0; i < 8; ++i) {
        loc[i] = (m[t * 8 + i] != 0) ? 1 : 0;
        c += loc[i];
    }
    s[t] = c;
    __syncthreads();

    // Inclusive Hillis-Steele scan over the 1024 per-thread counts.
#pragma unroll
    for (int off = 1; off < 1024; off <<= 1) {
        int add = (t >= off) ? s[t - off] : 0;
        __syncthreads();
        s[t] += add;
        __syncthreads();
    }
    const int incl  = s[t];
    const int total = s[1023];          // num_tokens[b]
    int run = incl - c;                  // exclusive count of selected before my chunk

    const int l0 = t * 8;
#pragma unroll
    for (int i = 0; i < 8; ++i) {
        const int l   = l0 + i;
        const int pos = loc[i] ? run : (total + (l - run));
        perm[(size_t)b * LL + pos] = l;
        run += loc[i];
    }
    if (t == 0) numtok[b] = total;
}

// ---------------------------------------------------------------------------
// Kernel 2: packed row gather through the CDNA5 async global<->LDS DMA path.
// Grid: (j-groups, B). 256 threads/block; one iteration moves 4 rows (16 KB).
// blockIdx.y carries the batch -> no runtime div/mod, numtok[b] hoisted.
// Double-buffered slot sets (it&1): iteration i's async stores remain in
// flight while iteration i+1 loads into the other set; the s_wait_asynccnt 0
// between i+1's load group and store group drains i's stores, so slot reuse
// at i+2 is safe. S_ENDPGM's implicit wait-idle covers the final stores.
// ---------------------------------------------------------------------------
__global__ __launch_bounds__(256)
void ChunkLayer_gather_rows(const float* __restrict__ hidden,
                            const int* __restrict__ perm,
                            const int* __restrict__ numtok,
                            float* __restrict__ out_h,
                            float* __restrict__ out_m,
                            int S) {
    __shared__ float buf[2 * ROWS_PER_ITER * DD];   // 32 KB staging
    const int t = threadIdx.x;
    const int b = blockIdx.y;
    // Low 32 bits of the generic address of an LDS object == LDS byte offset
    const unsigned lds_base = (unsigned)(size_t)(&buf[0]) + (unsigned)(t * 16);

    const float* hidden_b = hidden + (size_t)b * LL * DD;
    const int*   perm_b   = perm   + (size_t)b * LL;
    float*       out_h_b  = out_h  + (size_t)b * S * DD;
    float*       out_m_b  = out_m  + (size_t)b * S;
    const int    nt       = numtok[b];

    const int ngroups = (S + ROWS_PER_ITER - 1) / ROWS_PER_ITER;

    int it = 0;
    for (int g = blockIdx.x; g < ngroups; g += gridDim.x, ++it) {
        const int j0 = g * ROWS_PER_ITER;
        const unsigned set =
            lds_base + (unsigned)((it & 1) * ROWS_PER_ITER * ROW_BYTES);

        // ---- issue async gather loads (global -> LDS), 4 rows deep ----
#pragma unroll
        for (int k = 0; k < ROWS_PER_ITER; ++k) {
            const int j = j0 + k;
            if (j < S) {
                const int src_l = perm_b[j];
                const float* src = hidden_b + (size_t)src_l * DD + t * 4;
                const unsigned slot = set + (unsigned)(k * ROW_BYTES);
                asm volatile("global_load_async_to_lds_b128 %0, %1, off"
                             :: "v"(slot), "v"(src) : "memory");
            }
        }

        // Drain: current loads landed in LDS; also drains the previous
        // iteration's stores (other slot set), enabling its reuse next time.
        asm volatile("s_wait_asynccnt 0x0" ::: "memory");

        // ---- issue async packed stores (LDS -> global); no trailing wait ----
#pragma unroll
        for (int k = 0; k < ROWS_PER_ITER; ++k) {
            const int j = j0 + k;
            if (j < S) {
                float* dst = out_h_b + (size_t)j * DD + t * 4;
                const unsigned slot = set + (unsigned)(k * ROW_BYTES);
                asm volatile("global_store_async_from_lds_b128 %0, %1, off"
                             :: "v"(dst), "v"(slot) : "memory");
                if (t == 0) out_m_b[j] = (j < nt) ? 1.0f : 0.0f;
            }
        }
    }
}

// ---------------------------------------------------------------------------
// Launcher.
//   d_in[0]: hidden_states  float32  [B, L, D]
//   d_in[1]: boundary_mask  int32    [B, L]   (bool -> int per harness rule)
//   d_out  : [ next_hidden_states (B*S*D f32) | next_mask (B*S f32 0/1) ]
//   S recovered from out_size = B*S*(D+1).
//   d_ws   : perm (B*L ints) + numtok (B ints)
// ---------------------------------------------------------------------------
extern "C" void kernel_launch(void* const* d_in, const int* in_sizes, int n_in,
                              void* d_out, int out_size, void* d_ws, size_t ws_size,
                              hipStream_t stream) {
    const float* hidden = (const float*)d_in[0];
    const int*   mask   = (const int*)d_in[1];

    const int S = out_size / (BB * (DD + 1));
    if (S <= 0) return;

    int* perm   = (int*)d_ws;
    int* numtok = perm + (size_t)BB * LL;

    float* out_h = (float*)d_out;
    float* out_m = out_h + (size_t)BB * S * DD;

    ChunkLayer_scan_scatter<<<BB, 1024, 0, stream>>>(mask, perm, numtok);

    const int ngroups = (S + ROWS_PER_ITER - 1) / ROWS_PER_ITER;
    const int gx = ngroups < 8192 ? ngroups : 8192;
    dim3 grid(gx, BB);
    ChunkLayer_gather_rows<<<grid, 256, 0, stream>>>(hidden, perm, numtok,
                                                     out_h, out_m, S);
}